// AttnDecoderRNN_15350213116625
// MI455X (gfx1250) — compile-verified
//
#include <hip/hip_runtime.h>
#include <hip/hip_bf16.h>

// ---------------------------------------------------------------------------
// AttnDecoderRNN single step, H=2048, L=350, batch 1.
// Memory-bound GEMV chain (~143 MB fp32 weights -> ~6.2us @ 23.3 TB/s HBM).
// GEMVs: V_WMMA_F32_16X16X4_F32, 16 weight rows per wave, K advanced 4/step.
// Input vector staged into LDS by the Tensor Data Mover (tensor_load_to_lds,
// s_wait_tensorcnt) instead of per-thread staging loads.
// ---------------------------------------------------------------------------

#define H_DIM 2048
#define L_DIM 350

typedef unsigned int u32;
typedef __attribute__((ext_vector_type(2))) float v2f;
typedef __attribute__((ext_vector_type(8))) float v8f;
typedef __attribute__((ext_vector_type(4))) u32   v4u;
typedef __attribute__((ext_vector_type(8))) int   v8i;
typedef __attribute__((ext_vector_type(4))) int   v4i;

// Issue one TDM 1-D copy: nelem fp32 from `src` (global) to LDS byte offset
// `lds_byte_off`. Descriptor packing per CDNA5 ISA ch.8 (group0/group1),
// groups 2-3 zero (<=2D tensor). TENSORcnt tracks completion.
// This toolchain exposes the 6-arg builtin:
//   (v4u g0, v8i g1, v4i g2, v4i g3, v8i extra, i32 cpol)
__device__ __forceinline__ void tdm_load_vec(const float* src, u32 lds_byte_off,
                                             int nelem) {
    unsigned long long ga = (unsigned long long)(uintptr_t)src;
    v4u g0;
    g0[0] = 1u;                                   // count=1, user descriptor
    g0[1] = lds_byte_off;                         // lds_addr (bytes)
    g0[2] = (u32)ga;                              // global_addr[31:0]
    g0[3] = ((u32)(ga >> 32) & 0x01FFFFFFu)       // global_addr[56:32]
            | (2u << 30);                         // type=2 ("image")
    v8i g1;
    g1[0] = 0x00020000;                           // data_size=2 (4B), mask=0
    g1[1] = (int)(((u32)nelem & 0xFFFFu) << 16);  // tensor_dim0[15:0]
    g1[2] = (int)((((u32)nelem >> 16) & 0xFFFFu)  // tensor_dim0[31:16]
            | (1u << 16));                        // tensor_dim1 = 1
    g1[3] = (int)(((u32)nelem & 0xFFFFu) << 16);  // tile_dim0 = nelem
    g1[4] = 1;                                    // tile_dim1 = 1, tile_dim2 = 0
    g1[5] = nelem;                                // tensor_dim0_stride lo32
    g1[6] = 0;                                    // stride hi / dim1_stride lo
    g1[7] = 0;
    v4i zero4 = {0, 0, 0, 0};
    v8i zero8 = {0, 0, 0, 0, 0, 0, 0, 0};
    __builtin_amdgcn_tensor_load_to_lds(g0, g1, zero4, zero4, zero8,
                                        /*cpol=*/0);
}

// y[r] = act( bias[r] + dot(W[r,:], cat) ),  cat[k] = k<split ? v0[k] : v1[k-split]
// One wave = one 16-row tile; 8 waves (256 thr) per block; dyn-LDS = K floats.
template <int K, int ACT>
__global__ __launch_bounds__(256) void gemv_wmma(
    const float* __restrict__ W, const float* __restrict__ bias,
    const float* __restrict__ v0, const float* __restrict__ v1,
    int split, int N, float* __restrict__ out)
{
    extern __shared__ float smem[];

    // Wave 0 DMAs the (concatenated) input vector into LDS via the TDM.
    // LDS byte offset = low 32 bits of the generic pointer (LDS aperture
    // keeps the offset in the low word).
    if (threadIdx.x < 32) {
        const u32 lds_base = (u32)(uintptr_t)smem;
        tdm_load_vec(v0, lds_base, split);
        if (K > split) tdm_load_vec(v1, lds_base + 4u * (u32)split, K - split);
        __builtin_amdgcn_s_wait_tensorcnt(0);
    }
    __syncthreads();

    const int tid  = threadIdx.x;
    const int wave = tid >> 5;
    const int lane = tid & 31;
    const int half = lane >> 4;     // 0: lanes 0-15 (K=0,1), 1: lanes 16-31 (K=2,3)
    const int m    = lane & 15;     // row within tile for the A operand
    const int tile = blockIdx.x * 8 + wave;
    const int row0 = tile * 16;
    if (row0 >= N) return;          // wave-uniform; EXEC stays all-1s below

    int row = row0 + m;
    if (row >= N) row = N - 1;      // clamp loads for the ragged last tile
    const float* __restrict__ wrow = W + (size_t)row * (size_t)K;

    v8f c = {0.f, 0.f, 0.f, 0.f, 0.f, 0.f, 0.f, 0.f};

#pragma unroll 8
    for (int kb = 0; kb < K; kb += 4) {
        if ((kb & 31) == 0)  // one speculative prefetch per 128B of the row
            __builtin_prefetch(wrow + kb + 256, 0, 1);
        // A: 16x4 f32 tile. Each lane loads 4 consecutive row floats (b128),
        // then selects (k0,k1) for lanes 0-15 and (k2,k3) for lanes 16-31.
        const float4 a4 = *reinterpret_cast<const float4*>(wrow + kb);
        v2f a, b;
        a[0] = half ? a4.z : a4.x;
        a[1] = half ? a4.w : a4.y;
        // B: 4x16 f32 with every column equal to x[kb..kb+3]:
        // lanes 0-15 hold (x[kb],x[kb+1]), lanes 16-31 hold (x[kb+2],x[kb+3]).
        const float2 bv = reinterpret_cast<const float2*>(smem + kb)[half];
        b[0] = bv.x;
        b[1] = bv.y;
        c = __builtin_amdgcn_wmma_f32_16x16x4_f32(
                /*neg_a=*/false, a, /*neg_b=*/false, b,
                /*c_mod=*/(short)0, c, /*reuse_a=*/false, /*reuse_b=*/false);
    }

    // All D columns are identical: lane 0 holds rows M=0..7, lane 16 M=8..15.
    if (m == 0) {
        const int rbase = row0 + 8 * half;
#pragma unroll
        for (int i = 0; i < 8; ++i) {
            const int r = rbase + i;
            if (r < N) {
                float val = c[i] + bias[r];
                if (ACT) val = fmaxf(val, 0.f);
                out[r] = val;
            }
        }
    }
}

// softmax over n=350 logits; writes normalized weights to two destinations.
__global__ __launch_bounds__(512) void softmax_kernel(
    const float* __restrict__ logits, float* __restrict__ dst_ws,
    float* __restrict__ dst_out, int n)
{
    __shared__ float red[512];
    const int t = threadIdx.x;
    const float v = (t < n) ? logits[t] : -3.402823466e38f;

    red[t] = v;
    __syncthreads();
    for (int s = 256; s > 0; s >>= 1) {
        if (t < s) red[t] = fmaxf(red[t], red[t + s]);
        __syncthreads();
    }
    const float mx = red[0];
    __syncthreads();

    const float e = (t < n) ? expf(v - mx) : 0.f;
    red[t] = e;
    __syncthreads();
    for (int s = 256; s > 0; s >>= 1) {
        if (t < s) red[t] += red[t + s];
        __syncthreads();
    }
    const float inv = 1.0f / red[0];

    if (t < n) {
        const float w = e * inv;
        dst_ws[t]  = w;
        dst_out[t] = w;
    }
}

// context[j] = sum_l aw[l] * E[l][j]  (coalesced over j; aw cached in LDS)
__global__ __launch_bounds__(256) void context_kernel(
    const float* __restrict__ aw, const float* __restrict__ E,
    float* __restrict__ ctx)
{
    __shared__ float w[L_DIM];
    const int tid = threadIdx.x;
    for (int l = tid; l < L_DIM; l += 256) w[l] = aw[l];
    __syncthreads();

    const int j = blockIdx.x * 256 + tid;
    float acc = 0.f;
#pragma unroll 2
    for (int l = 0; l < L_DIM; ++l)
        acc = fmaf(w[l], E[(size_t)l * H_DIM + j], acc);
    ctx[j] = acc;
}

// Fused GRU gates: r,z,n then h_new; writes output and hidden slots of d_out.
__global__ __launch_bounds__(256) void gru_gates_kernel(
    const float* __restrict__ gi, const float* __restrict__ gh,
    const float* __restrict__ h, float* __restrict__ out)
{
    const int i = blockIdx.x * 256 + threadIdx.x;
    const float r  = 1.f / (1.f + expf(-(gi[i] + gh[i])));
    const float z  = 1.f / (1.f + expf(-(gi[H_DIM + i] + gh[H_DIM + i])));
    const float nn = tanhf(gi[2 * H_DIM + i] + r * gh[2 * H_DIM + i]);
    const float hn = (1.f - z) * nn + z * h[i];
    out[i]         = hn;   // output (1,1,H)
    out[H_DIM + i] = hn;   // hidden (1,1,H)
}

extern "C" void kernel_launch(void* const* d_in, const int* in_sizes, int n_in,
                              void* d_out, int out_size, void* d_ws, size_t ws_size,
                              hipStream_t stream) {
    const float* x   = (const float*)d_in[0];   // input  (1,1,H)
    const float* h   = (const float*)d_in[1];   // hidden (1,1,H)
    const float* E   = (const float*)d_in[2];   // encoder_outputs (L,H)
    const float* aW  = (const float*)d_in[3];   // attn_W (L, 2H)
    const float* ab  = (const float*)d_in[4];   // attn_b (L)
    const float* cW  = (const float*)d_in[5];   // comb_W (H, 2H)
    const float* cb  = (const float*)d_in[6];   // comb_b (H)
    const float* wih = (const float*)d_in[7];   // w_ih (3H, H)
    const float* whh = (const float*)d_in[8];   // w_hh (3H, H)
    const float* bih = (const float*)d_in[9];   // b_ih (3H)
    const float* bhh = (const float*)d_in[10];  // b_hh (3H)
    float* out = (float*)d_out;                 // [output(2048) | hidden(2048) | attn_w(350)]

    float* ws     = (float*)d_ws;
    float* logits = ws;            // 350 (padded 384)
    float* awts   = ws + 384;      // 350 (padded 384)
    float* ctx    = ws + 768;      // 2048
    float* g      = ws + 2816;     // 2048
    float* gi     = ws + 4864;     // 6144
    float* gh     = ws + 11008;    // 6144

    // 1) attention logits: (350 x 4096) @ [x;h]
    gemv_wmma<2 * H_DIM, 0><<<3, 256, 2 * H_DIM * sizeof(float), stream>>>(
        aW, ab, x, h, /*split=*/H_DIM, /*N=*/L_DIM, logits);
    // 2) softmax -> attn weights (also the 3rd reference output)
    softmax_kernel<<<1, 512, 0, stream>>>(logits, awts, out + 2 * H_DIM, L_DIM);
    // 3) context = attn_weights @ encoder_outputs
    context_kernel<<<H_DIM / 256, 256, 0, stream>>>(awts, E, ctx);
    // 4) g = relu((2048 x 4096) @ [x;ctx] + b)
    gemv_wmma<2 * H_DIM, 1><<<16, 256, 2 * H_DIM * sizeof(float), stream>>>(
        cW, cb, x, ctx, /*split=*/H_DIM, /*N=*/H_DIM, g);
    // 5) GRU input/hidden projections: (6144 x 2048) GEMVs
    gemv_wmma<H_DIM, 0><<<48, 256, H_DIM * sizeof(float), stream>>>(
        wih, bih, g, g, /*split=*/H_DIM, /*N=*/3 * H_DIM, gi);
    gemv_wmma<H_DIM, 0><<<48, 256, H_DIM * sizeof(float), stream>>>(
        whh, bhh, h, h, /*split=*/H_DIM, /*N=*/3 * H_DIM, gh);
    // 6) fused gates -> h_new -> output & hidden
    gru_gates_kernel<<<H_DIM / 256, 256, 0, stream>>>(gi, gh, h, out);
}